// ExpansionContrastModule_7352984011501
// MI455X (gfx1250) — compile-verified
//
#include <hip/hip_runtime.h>
#include <math.h>

typedef __attribute__((ext_vector_type(2))) float v2f;
typedef __attribute__((ext_vector_type(8))) float v8f;

#define B_  4
#define C_  128
#define CQ  32
#define H_  384
#define W_  384
#define HW  (H_ * W_)

// ---------------------------------------------------------------------------
// Kernel 1: x[b][m][q] = sum_k Win[m][k] * cen[b][k][q] + bin[m]
// GEMM via V_WMMA_F32_16X16X4_F32.  One wave -> 32(M) x 16(N pixels) tile.
// Block = 256 threads = 8 waves -> 128 pixels per block.
// A-frag layout (ISA 16x4 f32): VGPR0 = K0 (lanes 0-15) / K2 (lanes 16-31),
//                               VGPR1 = K1 / K3.  B-frag mirrors (K rows
// striped across lane halves, N = lane%16).
// ---------------------------------------------------------------------------
__global__ __launch_bounds__(256) void in_conv_wmma(
    const float* __restrict__ cen, const float* __restrict__ Win,
    const float* __restrict__ bin, float* __restrict__ x)
{
    const int wave = threadIdx.x >> 5;
    const int lane = threadIdx.x & 31;
    const int ln   = lane & 15;
    const int hi   = lane >> 4;          // 0 or 1 (lane half)

    const long p0 = (long)blockIdx.x * 128 + (long)wave * 16;
    const int  b  = (int)(p0 / HW);
    const int  q0 = (int)(p0 % HW);

    const float* cb = cen + (long)b * C_ * HW + q0 + ln;

    v8f acc0 = {};   // output channels m = 0..15
    v8f acc1 = {};   // output channels m = 16..31

    #pragma unroll 4
    for (int k = 0; k < C_; k += 4) {
        const int k0 = k + 2 * hi;       // K row for VGPR0 in this lane half
        v2f bf, a0, a1;
        bf.x = cb[(long)k0       * HW];
        bf.y = cb[(long)(k0 + 1) * HW];
        a0.x = Win[ln * C_ + k0];
        a0.y = Win[ln * C_ + k0 + 1];
        a1.x = Win[(16 + ln) * C_ + k0];
        a1.y = Win[(16 + ln) * C_ + k0 + 1];
        acc0 = __builtin_amdgcn_wmma_f32_16x16x4_f32(
                   false, a0, false, bf, (short)0, acc0, false, false);
        acc1 = __builtin_amdgcn_wmma_f32_16x16x4_f32(
                   false, a1, false, bf, (short)0, acc1, false, false);
    }

    // D layout: VGPR r holds M = r (lanes 0-15) / M = r+8 (lanes 16-31), N = ln
    float* xb = x + (long)b * CQ * HW + q0 + ln;
    #pragma unroll
    for (int r = 0; r < 8; ++r) {
        const int m = r + 8 * hi;
        xb[(long)m        * HW] = acc0[r] + bin[m];
        xb[(long)(16 + m) * HW] = acc1[r] + bin[16 + m];
    }
}

// ---------------------------------------------------------------------------
// Kernel 2: fused contrast stencil + out_conv(BN,SiLU) + sigmoid + final scale
// One thread per pixel; 32-channel feat held in registers; x is L2-resident.
// ---------------------------------------------------------------------------
__global__ __launch_bounds__(256) void contrast_fused(
    const float* __restrict__ x,   const float* __restrict__ cen,
    const float* __restrict__ mas, const float* __restrict__ scales,
    const float* __restrict__ bw,  const float* __restrict__ gamma,
    const float* __restrict__ beta,const float* __restrict__ mean,
    const float* __restrict__ var, const float* __restrict__ ow,
    const float* __restrict__ ob,  float* __restrict__ out)
{
    const long p = (long)blockIdx.x * blockDim.x + threadIdx.x;
    const int  b = (int)(p / HW);
    const int  q = (int)(p % HW);
    const int  h = q / W_;
    const int  w = q % W_;

    // o1_k = x(p) - x(p + off_k*d), o2_k = x(p) - x(p - off_k*d), zero-padded
    const int dy[4] = {-1, -1, -1, 0};
    const int dx[4] = {-1,  0,  1, -1};

    auto tap = [&](const float* base, int hh, int ww) -> float {
        return (hh >= 0 && hh < H_ && ww >= 0 && ww < W_) ? base[hh * W_ + ww]
                                                          : 0.0f;
    };

    float feat[CQ];
    for (int c = 0; c < CQ; ++c) {
        const float* xc  = x + ((long)(b * CQ + c)) * HW;
        const float  ctr = xc[q];
        float os[2];
        #pragma unroll
        for (int si = 0; si < 2; ++si) {
            const int d = si ? 3 : 1;     // SHIFTS = (1, 3)
            float mn = 0.0f;
            #pragma unroll
            for (int k4 = 0; k4 < 4; ++k4) {
                const float o1 = ctr - tap(xc, h + dy[k4] * d, w + dx[k4] * d);
                const float o2 = ctr - tap(xc, h - dy[k4] * d, w - dx[k4] * d);
                const float pr = o1 * o2;
                mn = (k4 == 0) ? pr : fminf(mn, pr);
            }
            os[si] = mn;
        }
        const float fx = fmaxf(os[0], os[1]) + 0.5f * (os[0] + os[1]);
        feat[c] = fx > 0.0f ? fx : 0.0f;  // relu(max + mean)
    }

    // out_conv: 1x1 (base_w) -> BN -> SiLU -> 1x1 (out_w) -> sigmoid
    float att_acc = ob[0];
    #pragma unroll 4
    for (int o = 0; o < CQ; ++o) {
        float y = 0.0f;
        const float* wr = bw + o * CQ;
        #pragma unroll
        for (int c2 = 0; c2 < CQ; ++c2) y = fmaf(wr[c2], feat[c2], y);
        const float inv = rsqrtf(var[o] + 1e-3f);
        y = (y - mean[o]) * (gamma[o] * inv) + beta[o];
        const float sig = 1.0f / (1.0f + __expf(-y));
        att_acc = fmaf(ow[o], y * sig, att_acc);
    }
    const float att = 1.0f / (1.0f + __expf(-att_acc));

    // softmax over the 4 scales (tiny, recomputed per thread; all cached)
    const float s0v = scales[0], s1v = scales[1];
    const float s2v = scales[2], s3v = scales[3];
    const float mx  = fmaxf(fmaxf(s0v, s1v), fmaxf(s2v, s3v));
    const float e0 = __expf(s0v - mx), e1 = __expf(s1v - mx);
    const float e2 = __expf(s2v - mx), e3 = __expf(s3v - mx);
    const float esum = e0 + e1 + e2 + e3;

    const float m = 1.0f / (1.0f + __expf(-mas[p]));   // mas is (B,1,H,W)
    const float factor =
        (att * m * e0 + m * e1 + e2 * att + e3) / esum;

    const float* cb   = cen + (long)b * C_ * HW + q;
    float*       outb = out + (long)b * C_ * HW + q;
    __builtin_prefetch(cb, 0, 1);               // -> global_prefetch_b8
    #pragma unroll 4
    for (int ch = 0; ch < C_; ++ch)
        outb[(long)ch * HW] = cb[(long)ch * HW] * factor;
}

// ---------------------------------------------------------------------------
extern "C" void kernel_launch(void* const* d_in, const int* in_sizes, int n_in,
                              void* d_out, int out_size, void* d_ws,
                              size_t ws_size, hipStream_t stream)
{
    const float* cen       = (const float*)d_in[0];
    const float* mas       = (const float*)d_in[1];
    const float* scales    = (const float*)d_in[2];
    const float* in_conv_w = (const float*)d_in[3];
    const float* in_conv_b = (const float*)d_in[4];
    const float* base_w    = (const float*)d_in[5];
    const float* bn_gamma  = (const float*)d_in[6];
    const float* bn_beta   = (const float*)d_in[7];
    const float* bn_mean   = (const float*)d_in[8];
    const float* bn_var    = (const float*)d_in[9];
    const float* out_w     = (const float*)d_in[10];
    const float* out_b     = (const float*)d_in[11];
    float*       out       = (float*)d_out;
    float*       x         = (float*)d_ws;   // 4*32*384*384*4 B = 75.5 MB

    const long P = (long)B_ * HW;            // 589824 pixels

    // K1: 128 pixels per block -> 4608 blocks of 256 threads
    in_conv_wmma<<<(int)(P / 128), 256, 0, stream>>>(cen, in_conv_w,
                                                     in_conv_b, x);
    // K2: one thread per pixel -> 2304 blocks of 256 threads
    contrast_fused<<<(int)(P / 256), 256, 0, stream>>>(
        x, cen, mas, scales, base_w, bn_gamma, bn_beta, bn_mean, bn_var,
        out_w, out_b, out);
}